// Tacotron_76390288326714
// MI455X (gfx1250) — compile-verified
//
#include <hip/hip_runtime.h>
#include <stdint.h>

#define B_    128
#define T_    256
#define ENC_  256
#define MEL_  80
#define STEPS_ 120
#define QD_   256
#define AD_   128
#define PRE1_ 256
#define PRE2_ 128
#define DEC_  256
#define MELR_ 560   // 80*7

typedef __attribute__((ext_vector_type(2))) float v2f;
typedef __attribute__((ext_vector_type(8))) float v8f;

// C[16,16] tile accumulate: A (LDS or global, row-major, stride lda),
// Bt = W^T row-major [K,N]. Per ISA: A lanes 0-15 = {K=k,k+1}, 16-31 = {k+2,k+3};
// B mirrors; D vgpr i = rows {i, 8+i}.
__device__ __forceinline__ v8f wmma_f32(v8f acc, const float* __restrict__ A, int lda,
                                        const float* __restrict__ Bt, int N, int K,
                                        int n0, int l, int kk) {
  for (int k = 0; k < K; k += 4) {
    v2f a, b;
    a.x = A[l * lda + k + kk];
    a.y = A[l * lda + k + kk + 1];
    b.x = Bt[(size_t)(k + kk) * N + n0 + l];
    b.y = Bt[(size_t)(k + kk + 1) * N + n0 + l];
    acc = __builtin_amdgcn_wmma_f32_16x16x4_f32(false, a, false, b, (short)0, acc,
                                                false, false);
  }
  return acc;
}

__device__ __forceinline__ float sigmoidf_(float x) { return 1.f / (1.f + __expf(-x)); }

__global__ void transpose_kernel(const float* __restrict__ src, float* __restrict__ dst,
                                 int N, int K) {
  int idx = blockIdx.x * blockDim.x + threadIdx.x;
  if (idx < N * K) {
    int n = idx / K, k = idx % K;
    dst[(size_t)k * N + n] = src[(size_t)n * K + k];
  }
}

// proc[b,t,:] = inputs[b,t,:] @ att_wi^T + att_bi   (rows = B*T = 32768, K=256, N=128)
__global__ __launch_bounds__(256) void proc_kernel(const float* __restrict__ inputs,
                                                   const float* __restrict__ wt,
                                                   const float* __restrict__ bias,
                                                   float* __restrict__ proc) {
  int wv = threadIdx.x >> 5, lane = threadIdx.x & 31;
  int l = lane & 15, hl = lane >> 4, kk = hl << 1;
  int tile = blockIdx.x * 8 + wv;      // 2048 blocks * 8 waves = 16384 tiles
  int rt = tile >> 3, ct = tile & 7;   // 2048 row-tiles x 8 col-tiles
  const float* A = inputs + (size_t)rt * 16 * ENC_;
  v8f acc = {};
  acc = wmma_f32(acc, A, ENC_, wt, AD_, ENC_, ct * 16, l, kk);
  float bb = bias[ct * 16 + l];
#pragma unroll
  for (int i = 0; i < 8; ++i)
    proc[((size_t)rt * 16 + hl * 8 + i) * AD_ + ct * 16 + l] = acc[i] + bb;
}

__global__ __launch_bounds__(512, 1) void tacotron_decoder(
    const float* __restrict__ inputs, const float* __restrict__ memory,
    const uint8_t* __restrict__ mask,
    const float* __restrict__ wt_pre1, const float* __restrict__ pre_b1,
    const float* __restrict__ wt_pre2, const float* __restrict__ pre_b2,
    const float* __restrict__ wt_agih, const float* __restrict__ wt_aghh,
    const float* __restrict__ agru_bih, const float* __restrict__ agru_bhh,
    const float* __restrict__ wt_attq, const float* __restrict__ att_bq,
    const float* __restrict__ att_v, const float* __restrict__ att_bv,
    const float* __restrict__ proc,
    const float* __restrict__ wt_proj, const float* __restrict__ proj_b,
    const float* __restrict__ wt_mel, const float* __restrict__ mel_b,
    const float* __restrict__ stop_w, const float* __restrict__ stop_b,
    float* __restrict__ outp, float* __restrict__ attp, float* __restrict__ stopp) {
  // Row strides padded by 4 floats -> conflict-free half-wave column access.
  __shared__ float sm[60160];
  float* A0  = sm;           // 16 x 84   mem_in
  float* PM1 = sm + 1344;    // 16 x 260  prenet1
  float* XB  = sm + 5504;    // 16 x 388  [pm(128) | ctx(256)]  (ctx = carried state)
  float* HB  = sm + 11712;   // 16 x 260  attn_h (carried state)
  float* GI  = sm + 15872;   // 16 x 772  gi
  float* GH  = sm + 28224;   // 16 x 772  gh
  float* PQ  = sm + 40576;   // 16 x 132  processed query
  float* SC  = sm + 42688;   // 16 x 260  score -> alpha
  float* DE  = sm + 46848;   // 16 x 260  dec_in
  float* OB  = sm + 51008;   // 16 x 564  mel out
  float* VB  = sm + 60032;   // 128       att_v cache

  const int tid = threadIdx.x;
  const int wv = tid >> 5, lane = tid & 31;
  const int l = lane & 15, hl = lane >> 4, kk = hl << 1;
  const int b0 = blockIdx.x * 16;

  for (int i = tid; i < 16 * 260; i += 512) HB[i] = 0.f;
  for (int i = tid; i < 16 * 388; i += 512) XB[i] = 0.f;
  for (int i = tid; i < 128; i += 512) VB[i] = att_v[i];
  __syncthreads();

  for (int s = 0; s < STEPS_; ++s) {
    // ---- A: stage teacher-forced mel frame (memory[b, (s-1)*7, :], zeros at s=0)
    for (int i = tid; i < 16 * MEL_; i += 512) {
      int r = i / MEL_, k = i % MEL_;
      A0[r * 84 + k] =
          (s == 0) ? 0.f
                   : memory[((size_t)(b0 + r) * 840 + (size_t)(s - 1) * 7) * MEL_ + k];
    }
    __syncthreads();

    // ---- B: prenet1 = relu(A0 @ pre_w1^T + b1), N=256 -> 16 tiles, 1/wave
    {
      int n0 = wv * 16;
      v8f acc = {};
      acc = wmma_f32(acc, A0, 84, wt_pre1, PRE1_, MEL_, n0, l, kk);
      float bb = pre_b1[n0 + l];
#pragma unroll
      for (int i = 0; i < 8; ++i) {
        float v = acc[i] + bb;
        PM1[(hl * 8 + i) * 260 + n0 + l] = v > 0.f ? v : 0.f;
      }
    }
    __syncthreads();

    // ---- C: prenet2 (8 tiles, -> XB[:,0:128]) + gh = h@whh^T+bhh (48 tiles), fused
    for (int tt = wv; tt < 56; tt += 16) {
      if (tt < 8) {
        int n0 = tt * 16;
        v8f acc = {};
        acc = wmma_f32(acc, PM1, 260, wt_pre2, PRE2_, PRE1_, n0, l, kk);
        float bb = pre_b2[n0 + l];
#pragma unroll
        for (int i = 0; i < 8; ++i) {
          float v = acc[i] + bb;
          XB[(hl * 8 + i) * 388 + n0 + l] = v > 0.f ? v : 0.f;
        }
      } else {
        int n0 = (tt - 8) * 16;
        v8f acc = {};
        acc = wmma_f32(acc, HB, 260, wt_aghh, 768, QD_, n0, l, kk);
        float bb = agru_bhh[n0 + l];
#pragma unroll
        for (int i = 0; i < 8; ++i) GH[(hl * 8 + i) * 772 + n0 + l] = acc[i] + bb;
      }
    }
    __syncthreads();

    // ---- D: gi = [pm|ctx] @ wih^T + bih  (K=384, 48 tiles)
    for (int tt = wv; tt < 48; tt += 16) {
      int n0 = tt * 16;
      v8f acc = {};
      acc = wmma_f32(acc, XB, 388, wt_agih, 768, 384, n0, l, kk);
      float bb = agru_bih[n0 + l];
#pragma unroll
      for (int i = 0; i < 8; ++i) GI[(hl * 8 + i) * 772 + n0 + l] = acc[i] + bb;
    }
    __syncthreads();

    // ---- E: GRU gate fuse -> new attn_h (in place, thread-private element)
    for (int i = tid; i < 4096; i += 512) {
      int r = i >> 8, j = i & 255;
      float ir = GI[r * 772 + j], iz = GI[r * 772 + 256 + j], in_ = GI[r * 772 + 512 + j];
      float hr = GH[r * 772 + j], hz = GH[r * 772 + 256 + j], hn = GH[r * 772 + 512 + j];
      float rg = sigmoidf_(ir + hr), zg = sigmoidf_(iz + hz);
      float ng = tanhf(in_ + rg * hn);
      HB[r * 260 + j] = (1.f - zg) * ng + zg * HB[r * 260 + j];
    }
    __syncthreads();

    // ---- F: pq = attn_h @ att_wq^T + bq (8 tiles)
    if (wv < 8) {
      int n0 = wv * 16;
      v8f acc = {};
      acc = wmma_f32(acc, HB, 260, wt_attq, AD_, QD_, n0, l, kk);
      float bb = att_bq[n0 + l];
#pragma unroll
      for (int i = 0; i < 8; ++i) PQ[(hl * 8 + i) * 132 + n0 + l] = acc[i] + bb;
    }
    __syncthreads();

    // ---- G: score[r,t] = sum_d tanh(pq[r,d] + proc[b,t,d]) * v[d] + bv ; mask
    {
      float bv = att_bv[0];
      for (int i = tid; i < 4096; i += 512) {
        int r = i >> 8, t = i & 255;
        const float* pr = proc + (((size_t)(b0 + r)) * T_ + t) * AD_;
        const float* qr = PQ + r * 132;
        float sacc = 0.f;
#pragma unroll 4
        for (int d = 0; d < AD_; ++d) sacc += tanhf(qr[d] + pr[d]) * VB[d];
        sacc += bv;
        if (!mask[(size_t)(b0 + r) * T_ + t]) sacc = -1e9f;
        SC[r * 260 + t] = sacc;
      }
    }
    __syncthreads();

    // ---- H: softmax over T, one row per wave (16 waves = 16 rows)
    {
      int r = wv;
      float vals[8];
      float mx = -3.4e38f;
#pragma unroll
      for (int i = 0; i < 8; ++i) {
        vals[i] = SC[r * 260 + lane + 32 * i];
        mx = fmaxf(mx, vals[i]);
      }
#pragma unroll
      for (int o = 16; o > 0; o >>= 1) mx = fmaxf(mx, __shfl_xor(mx, o, 32));
      float ssum = 0.f;
#pragma unroll
      for (int i = 0; i < 8; ++i) {
        vals[i] = __expf(vals[i] - mx);
        ssum += vals[i];
      }
#pragma unroll
      for (int o = 16; o > 0; o >>= 1) ssum += __shfl_xor(ssum, o, 32);
      float inv = 1.f / ssum;
#pragma unroll
      for (int i = 0; i < 8; ++i) SC[r * 260 + lane + 32 * i] = vals[i] * inv;
    }
    __syncthreads();

    // ---- I: ctx[r,d] = sum_t alpha[r,t] * inputs[b,t,d]  (batched -> VALU)
    for (int i = tid; i < 4096; i += 512) {
      int r = i >> 8, d = i & 255;
      const float* inp = inputs + ((size_t)(b0 + r)) * T_ * ENC_ + d;
      const float* al = SC + r * 260;
      float c = 0.f;
      for (int t = 0; t < T_; ++t) c += al[t] * inp[(size_t)t * ENC_];
      XB[r * 388 + 128 + d] = c;
    }
    __syncthreads();

    // ---- J: dec_in = [attn_h | ctx] @ proj_w^T + b  (K=512 split into 2x256)
    {
      int n0 = wv * 16;
      v8f acc = {};
      acc = wmma_f32(acc, HB, 260, wt_proj, DEC_, QD_, n0, l, kk);
      acc = wmma_f32(acc, XB + 128, 388, wt_proj + (size_t)256 * DEC_, DEC_, ENC_, n0, l, kk);
      float bb = proj_b[n0 + l];
#pragma unroll
      for (int i = 0; i < 8; ++i) DE[(hl * 8 + i) * 260 + n0 + l] = acc[i] + bb;
    }
    __syncthreads();
    // (decoder GRU stack dh0/dh1 is dead code: dec_out == dec_in in the reference)

    // ---- K: out = dec_in @ mel_w^T + mel_b (N=560, 35 tiles) -> LDS + global
    for (int tt = wv; tt < 35; tt += 16) {
      int n0 = tt * 16;
      v8f acc = {};
      acc = wmma_f32(acc, DE, 260, wt_mel, MELR_, DEC_, n0, l, kk);
      float bb = mel_b[n0 + l];
#pragma unroll
      for (int i = 0; i < 8; ++i) {
        int r = hl * 8 + i;
        float v = acc[i] + bb;
        OB[r * 564 + n0 + l] = v;
        outp[(((size_t)(b0 + r)) * STEPS_ + s) * MELR_ + n0 + l] = v;
      }
    }
    __syncthreads();

    // ---- L: write attentions; stop = [dec_in|out] . stop_w + b (one row/wave)
    for (int i = tid; i < 4096; i += 512) {
      int r = i >> 8, t = i & 255;
      attp[(((size_t)(b0 + r)) * STEPS_ + s) * T_ + t] = SC[r * 260 + t];
    }
    {
      int r = wv;
      float acc = 0.f;
      for (int j = lane; j < 816; j += 32) {
        float x = (j < 256) ? DE[r * 260 + j] : OB[r * 564 + (j - 256)];
        acc += stop_w[j] * x;
      }
#pragma unroll
      for (int o = 16; o > 0; o >>= 1) acc += __shfl_xor(acc, o, 32);
      if (lane == 0) stopp[((size_t)(b0 + r)) * STEPS_ + s] = acc + stop_b[0];
    }
    __syncthreads();
  }
}

extern "C" void kernel_launch(void* const* d_in, const int* in_sizes, int n_in,
                              void* d_out, int out_size, void* d_ws, size_t ws_size,
                              hipStream_t stream) {
  (void)in_sizes; (void)n_in; (void)out_size; (void)ws_size;
  const float*   inputs   = (const float*)d_in[0];
  const float*   memory   = (const float*)d_in[1];
  const uint8_t* mask     = (const uint8_t*)d_in[2];
  const float*   pre_w1   = (const float*)d_in[3];
  const float*   pre_b1   = (const float*)d_in[4];
  const float*   pre_w2   = (const float*)d_in[5];
  const float*   pre_b2   = (const float*)d_in[6];
  const float*   agru_wih = (const float*)d_in[7];
  const float*   agru_whh = (const float*)d_in[8];
  const float*   agru_bih = (const float*)d_in[9];
  const float*   agru_bhh = (const float*)d_in[10];
  const float*   att_wq   = (const float*)d_in[11];
  const float*   att_bq   = (const float*)d_in[12];
  const float*   att_wi   = (const float*)d_in[13];
  const float*   att_bi   = (const float*)d_in[14];
  const float*   att_v    = (const float*)d_in[15];
  const float*   att_bv   = (const float*)d_in[16];
  const float*   proj_w   = (const float*)d_in[17];
  const float*   proj_b   = (const float*)d_in[18];
  // d_in[19..26] = d0_*/d1_* decoder GRU weights: provably dead in the reference.
  const float*   mel_w    = (const float*)d_in[27];
  const float*   mel_b    = (const float*)d_in[28];
  const float*   stop_w   = (const float*)d_in[29];
  const float*   stop_b   = (const float*)d_in[30];

  float* ws = (float*)d_ws;
  float* proc    = ws;                 size_t off = (size_t)B_ * T_ * AD_;
  float* wt_pre1 = ws + off;           off += (size_t)MEL_ * PRE1_;
  float* wt_pre2 = ws + off;           off += (size_t)PRE1_ * PRE2_;
  float* wt_agih = ws + off;           off += (size_t)384 * 768;
  float* wt_aghh = ws + off;           off += (size_t)256 * 768;
  float* wt_attq = ws + off;           off += (size_t)QD_ * AD_;
  float* wt_atti = ws + off;           off += (size_t)ENC_ * AD_;
  float* wt_proj = ws + off;           off += (size_t)512 * DEC_;
  float* wt_mel  = ws + off;           off += (size_t)DEC_ * MELR_;

  auto tr = [&](const float* s, float* d, int N, int K) {
    int n = N * K;
    transpose_kernel<<<(n + 255) / 256, 256, 0, stream>>>(s, d, N, K);
  };
  tr(pre_w1,   wt_pre1, PRE1_, MEL_);   // [256,80]  -> [80,256]
  tr(pre_w2,   wt_pre2, PRE2_, PRE1_);  // [128,256] -> [256,128]
  tr(agru_wih, wt_agih, 768, 384);
  tr(agru_whh, wt_aghh, 768, 256);
  tr(att_wq,   wt_attq, AD_, QD_);
  tr(att_wi,   wt_atti, AD_, ENC_);
  tr(proj_w,   wt_proj, DEC_, 512);
  tr(mel_w,    wt_mel,  MELR_, DEC_);

  proc_kernel<<<2048, 256, 0, stream>>>(inputs, wt_atti, att_bi, proc);

  float* outp  = (float*)d_out;
  float* attp  = outp + (size_t)B_ * STEPS_ * MELR_;
  float* stopp = attp + (size_t)B_ * STEPS_ * T_;

  tacotron_decoder<<<8, 512, 0, stream>>>(
      inputs, memory, mask, wt_pre1, pre_b1, wt_pre2, pre_b2, wt_agih, wt_aghh,
      agru_bih, agru_bhh, wt_attq, att_bq, att_v, att_bv, proc, wt_proj, proj_b,
      wt_mel, mel_b, stop_w, stop_b, outp, attp, stopp);
}